// CrossInnerProductWithBuyer_25031069401581
// MI455X (gfx1250) — compile-verified
//
#include <hip/hip_runtime.h>

// B=16384, W=10, N=64, E=128 ; output [B, 75]
// rows: 0..9 windows, 10..73 -neg, 74 buy

typedef __attribute__((ext_vector_type(2))) float v2f;
typedef __attribute__((ext_vector_type(8))) float v8f;

#define WAVES_PER_BLOCK 4
#define ROWSTRIDE 132   // floats; 132 % 64 == 4 -> conflict-free ds_load_b64 pattern
#define E 128
#define NROWS 75

__global__ __launch_bounds__(WAVES_PER_BLOCK * 32)
void cross_inner_wmma_kernel(const float* __restrict__ cv,   // [B,128] center
                             const float* __restrict__ wv,   // [B,10,128] windows
                             const float* __restrict__ nv,   // [B,64,128] neg
                             const float* __restrict__ bv,   // [B,128] buy
                             float* __restrict__ out,        // [B,75]
                             int Btot) {
    // per-wave private LDS: 16 padded rows + center vector
    __shared__ float lds[WAVES_PER_BLOCK][16 * ROWSTRIDE + E];

    const int wave = threadIdx.x >> 5;
    const int lane = threadIdx.x & 31;
    const int b    = blockIdx.x * WAVES_PER_BLOCK + wave;   // wave-uniform
    if (b >= Btot) return;                                  // scalar branch, EXEC stays full

    float* tile = &lds[wave][0];
    float* ctr  = &lds[wave][16 * ROWSTRIDE];

    // ---- stage center vector into LDS (coalesced 512B) ----
    {
        const float4* src = (const float4*)(cv + (size_t)b * E);
        *(float4*)(ctr + lane * 4) = src[lane];
    }

    const int half = lane >> 4;   // 0: K pair {0,1}, 1: K pair {2,3}
    const int nrow = lane & 15;   // output column / row-in-tile
    const float* browp = tile + nrow * ROWSTRIDE;

    for (int t = 0; t < 5; ++t) {
        // ---- stage 16 rows (8KB) into LDS, coalesced B128 both ways ----
        #pragma unroll
        for (int r16 = 0; r16 < 16; ++r16) {
            const int r = t * 16 + r16;                 // wave-uniform global row
            const float* src;
            if (r < 10)      src = wv + ((size_t)b * 10 + r) * E;
            else if (r < 74) src = nv + ((size_t)b * 64 + (r - 10)) * E;
            else             src = bv + (size_t)b * E;  // r==74 buy; r>=75 pad (safe addr)
            float4 v = ((const float4*)src)[lane];
            *(float4*)(tile + r16 * ROWSTRIDE + lane * 4) = v;
        }

        // ---- K-loop: 32 x v_wmma_f32_16x16x4_f32 ----
        // A: all 16 rows = c (lanes0-15 read c[k..k+1], lanes16-31 read c[k+2..k+3] -> broadcast)
        // B: B[k][n] = tile_row_n[k0+k]  (lane layout: VGPR0 = K0|K2, VGPR1 = K1|K3)
        v8f acc = {0.f, 0.f, 0.f, 0.f, 0.f, 0.f, 0.f, 0.f};
        #pragma unroll
        for (int i = 0; i < 32; ++i) {
            const int k = 4 * i + 2 * half;
            v2f amat = *(const v2f*)(ctr + k);
            v2f bmat = *(const v2f*)(browp + k);
            acc = __builtin_amdgcn_wmma_f32_16x16x4_f32(
                false, amat, false, bmat, (short)0, acc, false, false);
        }

        // ---- D row0 (acc[0]): lane n holds dot(c, row n); lanes 16-31 are duplicates ----
        const int r = t * 16 + nrow;
        float val = acc[0];
        if (r >= 10 && r < 74) val = -val;              // out2 = -neg . center
        if (half == 0 && r < NROWS) out[(size_t)b * NROWS + r] = val;
    }
}

extern "C" void kernel_launch(void* const* d_in, const int* in_sizes, int n_in,
                              void* d_out, int out_size, void* d_ws, size_t ws_size,
                              hipStream_t stream) {
    (void)n_in; (void)d_ws; (void)ws_size; (void)out_size;
    const float* cv = (const float*)d_in[0];  // center_vec  [B,1,128]
    const float* wv = (const float*)d_in[1];  // windows_vecs [B,10,128]
    const float* nv = (const float*)d_in[2];  // neg_vecs    [B,64,128]
    const float* bv = (const float*)d_in[3];  // buy_vec     [B,1,128]
    float* out = (float*)d_out;               // [B,75]

    const int Btot = in_sizes[0] / E;         // 16384
    const int blocks = (Btot + WAVES_PER_BLOCK - 1) / WAVES_PER_BLOCK;
    cross_inner_wmma_kernel<<<blocks, WAVES_PER_BLOCK * 32, 0, stream>>>(
        cv, wv, nv, bv, out, Btot);
}